// FastLoRAEmbedding_11175504904544
// MI455X (gfx1250) — compile-verified
//
#include <hip/hip_runtime.h>

typedef float v2f __attribute__((ext_vector_type(2)));
typedef float v8f __attribute__((ext_vector_type(8)));

#define VOCAB_SZ 50257
#define DIM 2048
#define RLORA 64
#define LORA_SCALE 16.0f          // 128 / sqrt(64)

#define TPB 16                    // tokens per block (M-tile)
#define LDS_A_STRIDE 66           // 64 + 2 pad: avoids LDS bank conflicts, keeps 8B align

__global__ __launch_bounds__(256) void lora_embed_wmma_kernel(
    const int*   __restrict__ x,       // [n_tokens] token ids
    const float* __restrict__ weight,  // [VOCAB, DIM]
    const float* __restrict__ lora_A,  // [R, VOCAB]
    const float* __restrict__ lora_B,  // [DIM, R]
    float*       __restrict__ out,     // [n_tokens, DIM]
    int n_tokens)
{
    __shared__ float lds_a[TPB * LDS_A_STRIDE];  // scaled A-tile: [m][r]
    __shared__ int   lds_tok[TPB];

    const int tid      = threadIdx.x;
    const int tok_base = blockIdx.x * TPB;

    if (tid < TPB) {
        int t = tok_base + tid;
        lds_tok[tid] = (t < n_tokens) ? x[t] : 0;
    }
    __syncthreads();

    // Gather lora_A columns for this token tile, pre-scaled by alpha/sqrt(r).
    #pragma unroll
    for (int i = 0; i < 4; ++i) {
        int p = tid + i * 256;         // 0..1023
        int m = p & 15;                // token within tile
        int r = p >> 4;                // 0..63
        int tok = lds_tok[m];
        lds_a[m * LDS_A_STRIDE + r] = LORA_SCALE * lora_A[r * VOCAB_SZ + tok];
    }
    __syncthreads();

    const int wave  = tid >> 5;        // 0..7
    const int lane  = tid & 31;
    const int ncol  = lane & 15;       // N within 16-wide tile
    const int half  = lane >> 4;       // 0: lanes 0-15, 1: lanes 16-31
    const int koff  = half * 2;        // K sub-offset per WMMA layout

    // Stage the A fragments once: lane supplies row M = lane&15;
    // fragment kk holds K = 4kk+koff, 4kk+koff+1.  (16 x ds_load_b64)
    const float* aRow = &lds_a[(lane & 15) * LDS_A_STRIDE + koff];
    v2f afrag[RLORA / 4];
    #pragma unroll
    for (int kk = 0; kk < RLORA / 4; ++kk)
        afrag[kk] = *(const v2f*)(aRow + 4 * kk);

    // Per-lane row offsets for the 8 C/D rows this lane touches (M = v + 8*half).
    int woff[8], ooff[8];
    #pragma unroll
    for (int v = 0; v < 8; ++v) {
        int m = v + half * 8;
        woff[v] = lds_tok[m] * DIM;
        ooff[v] = (tok_base + m) * DIM;
    }
    const bool full = (tok_base + TPB) <= n_tokens;   // block-uniform

    if (full) {
        for (int nt = wave; nt < DIM / 16; nt += 8) {
            const int col = nt * 16 + ncol;
            const float* bCol = &lora_B[col * RLORA + koff];

            // Stage all 16 B fragments into distinct registers -> one load clause.
            v2f b[RLORA / 4];
            #pragma unroll
            for (int kk = 0; kk < RLORA / 4; ++kk)
                b[kk] = *(const v2f*)(bCol + 4 * kk);

            // Base-embedding gathers queue up behind the B loads (same LOADcnt,
            // in-order return) and complete under the WMMA chain.
            float base[8];
            #pragma unroll
            for (int v = 0; v < 8; ++v)
                base[v] = weight[woff[v] + col];

            // Keep the loads as an issue clause; don't interleave WMMAs upward.
            __builtin_amdgcn_sched_barrier(0);

            v8f acc = {0.f, 0.f, 0.f, 0.f, 0.f, 0.f, 0.f, 0.f};
            #pragma unroll
            for (int kk = 0; kk < RLORA / 4; ++kk)
                acc = __builtin_amdgcn_wmma_f32_16x16x4_f32(
                    false, afrag[kk], false, b[kk], (short)0, acc, false, false);

            #pragma unroll
            for (int v = 0; v < 8; ++v)
                __builtin_nontemporal_store(base[v] + acc[v], &out[ooff[v] + col]);
        }
    } else {
        for (int nt = wave; nt < DIM / 16; nt += 8) {
            const int col = nt * 16 + ncol;
            const float* bCol = &lora_B[col * RLORA + koff];

            v2f b[RLORA / 4];
            #pragma unroll
            for (int kk = 0; kk < RLORA / 4; ++kk)
                b[kk] = *(const v2f*)(bCol + 4 * kk);

            v8f acc = {0.f, 0.f, 0.f, 0.f, 0.f, 0.f, 0.f, 0.f};
            #pragma unroll
            for (int kk = 0; kk < RLORA / 4; ++kk)
                acc = __builtin_amdgcn_wmma_f32_16x16x4_f32(
                    false, afrag[kk], false, b[kk], (short)0, acc, false, false);

            #pragma unroll
            for (int v = 0; v < 8; ++v) {
                int m = v + half * 8;
                if (tok_base + m < n_tokens) {
                    float base = weight[woff[v] + col];
                    out[ooff[v] + col] = base + acc[v];
                }
            }
        }
    }
}

extern "C" void kernel_launch(void* const* d_in, const int* in_sizes, int n_in,
                              void* d_out, int out_size, void* d_ws, size_t ws_size,
                              hipStream_t stream) {
    const int*   x      = (const int*)  d_in[0];
    const float* weight = (const float*)d_in[1];
    const float* lora_A = (const float*)d_in[2];
    const float* lora_B = (const float*)d_in[3];
    float*       out    = (float*)d_out;

    const int n_tokens = in_sizes[0];            // B*S = 16384
    const int blocks   = (n_tokens + TPB - 1) / TPB;

    hipLaunchKernelGGL(lora_embed_wmma_kernel, dim3(blocks), dim3(256), 0, stream,
                       x, weight, lora_A, lora_B, out, n_tokens);
}